// MultiHeadSelfAttention_51445118271936
// MI455X (gfx1250) — compile-verified
//
#include <hip/hip_runtime.h>
#include <math.h>

typedef __attribute__((ext_vector_type(2))) float v2f;
typedef __attribute__((ext_vector_type(8))) float v8f;

#define T_SEQ   4096
#define D_MODEL 1024
#define NH      16
#define HDIM    64
#define WLOC    128
#define GSTR    256
#define KC      32
#define LDP     68          // padded LDS row stride (floats): 272B, 16B-aligned, conflict-free

// ---------------------------------------------------------------------------
// RoPE (interleaved pairs, reference's emb[::2] indexing):
//   theta_i = t * 10000^(-((4*i) mod 64)/64),  pair i within head
// ---------------------------------------------------------------------------
__global__ __launch_bounds__(256) void rope_kernel(const float* __restrict__ x,
                                                   float* __restrict__ xr) {
    int idx = blockIdx.x * blockDim.x + threadIdx.x;      // pair index in [0, T*D/2)
    int t   = idx / (D_MODEL / 2);
    int rem = idx - t * (D_MODEL / 2);
    int i   = rem & (HDIM / 2 - 1);                       // pair index within head
    int off = t * D_MODEL + (rem << 1);
    float x1 = x[off], x2 = x[off + 1];
    float e    = (float)((4 * i) & 63) * (1.0f / 64.0f);
    float invf = __powf(10000.0f, -e);
    float th   = (float)t * invf;
    float sv, cv;
    __sincosf(th, &sv, &cv);
    xr[off]     = x1 * cv - x2 * sv;
    xr[off + 1] = x1 * sv + x2 * cv;
}

// ---------------------------------------------------------------------------
// GEMM: C[M x N] = A[M x K] * W[N x K]^T + bias,  fp32 WMMA 16x16x4.
// One wave computes a 16x64 strip. K in chunks of 32: 40 fragment loads
// batched per chunk, then 32 WMMAs back-to-back (schedules with
// s_wait_loadcnt 0x2..0x5 and cross-chunk pipelining).
// ---------------------------------------------------------------------------
__global__ __launch_bounds__(32) void gemm_wmma(const float* __restrict__ A, int lda,
                                                const float* __restrict__ Wt,  // (N x K) row-major
                                                const float* __restrict__ bias,
                                                float* __restrict__ C, int ldc,
                                                int K) {
    const int lane = threadIdx.x;
    const int m    = lane & 15;
    const int hf   = lane >> 4;                 // lane half: K sub-offset
    const int ms   = blockIdx.x * 16;
    const int ns   = blockIdx.y * 64;

    v8f acc0 = {}, acc1 = {}, acc2 = {}, acc3 = {};

    const float* arow = A  + (size_t)(ms + m) * lda + 2 * hf;
    const float* b0   = Wt + (size_t)(ns +  0 + m) * K + 2 * hf;
    const float* b1   = Wt + (size_t)(ns + 16 + m) * K + 2 * hf;
    const float* b2   = Wt + (size_t)(ns + 32 + m) * K + 2 * hf;
    const float* b3   = Wt + (size_t)(ns + 48 + m) * K + 2 * hf;

    for (int kc = 0; kc < K; kc += KC) {
        v2f af[8], wf0[8], wf1[8], wf2[8], wf3[8];
#pragma unroll
        for (int s = 0; s < 8; ++s) {
            const int k = kc + 4 * s;
            af[s]  = *(const v2f*)(arow + k);
            wf0[s] = *(const v2f*)(b0 + k);
            wf1[s] = *(const v2f*)(b1 + k);
            wf2[s] = *(const v2f*)(b2 + k);
            wf3[s] = *(const v2f*)(b3 + k);
        }
#pragma unroll
        for (int s = 0; s < 8; ++s) {
            acc0 = __builtin_amdgcn_wmma_f32_16x16x4_f32(false, af[s], false, wf0[s], (short)0, acc0, false, false);
            acc1 = __builtin_amdgcn_wmma_f32_16x16x4_f32(false, af[s], false, wf1[s], (short)0, acc1, false, false);
            acc2 = __builtin_amdgcn_wmma_f32_16x16x4_f32(false, af[s], false, wf2[s], (short)0, acc2, false, false);
            acc3 = __builtin_amdgcn_wmma_f32_16x16x4_f32(false, af[s], false, wf3[s], (short)0, acc3, false, false);
        }
    }

#pragma unroll
    for (int v = 0; v < 8; ++v) {
        int row = ms + v + 8 * hf;
        float* crow = C + (size_t)row * ldc + ns;
        crow[ 0 + m] = acc0[v] + bias[ns +  0 + m];
        crow[16 + m] = acc1[v] + bias[ns + 16 + m];
        crow[32 + m] = acc2[v] + bias[ns + 32 + m];
        crow[48 + m] = acc3[v] + bias[ns + 48 + m];
    }
}

// ---------------------------------------------------------------------------
// Async DMA of one 16x64 f32 tile (row stride 3*D_MODEL in global) into LDS
// (row stride LDP floats), as 8 x global_load_async_to_lds_b128 per wave
// (coalesced 512B per issue). Tracked by ASYNCcnt.
// ---------------------------------------------------------------------------
__device__ __forceinline__ void async_tile_copy(const float* __restrict__ base,
                                                unsigned gbyte0, unsigned lbyte0,
                                                int lane) {
#pragma unroll
    for (int i = 0; i < 8; ++i) {
        const int idx = i * 32 + lane;
        const int row = idx >> 4;                 // 16 x b128 chunks per 64-float row
        const int ch  = idx & 15;
        const unsigned go = gbyte0 + (unsigned)(row * (3 * D_MODEL) + ch * 4) * 4u;
        const unsigned lo = lbyte0 + (unsigned)(row * LDP + ch * 4) * 4u;
        asm volatile("global_load_async_to_lds_b128 %0, %1, %2"
                     :: "v"(lo), "v"(go), "s"(base) : "memory");
    }
}

// ---------------------------------------------------------------------------
// Flash attention, one wave per (16-row Q tile, head). Sparse tile schedule
// (local window +/-8 tiles, global tiles every 16). K/V tiles double-buffered
// in LDS via async global->LDS DMA: tile kt+next streams while tile kt is
// computed; wave only waits s_wait_asynccnt 16. S and P*V via fp32 WMMA;
// P transposed C-layout -> A-layout via LDS.
// ---------------------------------------------------------------------------
__global__ __launch_bounds__(32) void attn_kernel(const float* __restrict__ qkv,
                                                  float* __restrict__ o) {
    __shared__ __align__(16) float Kt[2][16 * LDP];
    __shared__ __align__(16) float Vt[2][16 * LDP];
    __shared__ float Pt[16 * 17];

    const int lane = threadIdx.x;
    const int m    = lane & 15;
    const int hf   = lane >> 4;
    const int qt   = blockIdx.x;
    const int qs   = qt * 16;
    const int h    = blockIdx.y;
    const float scale = 0.125f;                 // 1/sqrt(64)
    const size_t ld = 3 * D_MODEL;

    // local-window tile range: kt in [qt-8, qt+8] clipped
    const int lo = (qt - 8 > 0) ? qt - 8 : 0;
    const int hi = (qt + 8 < T_SEQ / 16 - 1) ? qt + 8 : T_SEQ / 16 - 1;

    // Q tile in A-fragment layout: 16 steps x float2 per lane (one-time load)
    v2f qa[16];
    {
        const float* qrow = qkv + (size_t)h * HDIM + (size_t)(qs + m) * ld + 2 * hf;
#pragma unroll
        for (int s = 0; s < 16; ++s) qa[s] = *(const v2f*)(qrow + 4 * s);
    }

    v8f o0 = {}, o1 = {}, o2 = {}, o3 = {};
    float mrow[8], lrow[8];
#pragma unroll
    for (int v = 0; v < 8; ++v) { mrow[v] = -1e30f; lrow[v] = 0.0f; }

    // byte offsets of K / V panels for this head within qkv
    const unsigned koff = (unsigned)((D_MODEL     + h * HDIM) * 4);
    const unsigned voff = (unsigned)((2 * D_MODEL + h * HDIM) * 4);

    // prologue: start DMA of first tile (kt = 0 is always in the set)
    int kt = 0, buf = 0;
    async_tile_copy(qkv, koff + (unsigned)(0) * (unsigned)(ld * 4), (unsigned)(size_t)&Kt[0][0], lane);
    async_tile_copy(qkv, voff + (unsigned)(0) * (unsigned)(ld * 4), (unsigned)(size_t)&Vt[0][0], lane);

    while (kt >= 0) {
        // ---- next tile in the sparse schedule (closed form) ----
        int nm16 = (kt | 15) + 1;                          // next multiple of 16
        int nl   = (kt + 1 < lo) ? lo : kt + 1;            // next local tile
        if (nl > hi) nl = 1 << 30;
        int nxt = (nm16 < nl) ? nm16 : nl;
        if (nxt >= T_SEQ / 16) nxt = -1;

        // ---- start DMA of next tile into the other buffer, then wait for
        //      the oldest 16 issues (current tile) to complete ----
        if (nxt >= 0) {
            const unsigned gb = (unsigned)((size_t)nxt * 16u * (unsigned)(ld * 4));
            async_tile_copy(qkv, koff + gb, (unsigned)(size_t)&Kt[buf ^ 1][0], lane);
            async_tile_copy(qkv, voff + gb, (unsigned)(size_t)&Vt[buf ^ 1][0], lane);
            asm volatile("s_wait_asynccnt 0x10" ::: "memory");
        } else {
            asm volatile("s_wait_asynccnt 0x0" ::: "memory");
        }

        const int j0 = kt * 16;

        // ---- K fragments from LDS, then 16 WMMAs for S = Q*K^T ----
        v2f kf[16];
#pragma unroll
        for (int s = 0; s < 16; ++s) {
            kf[s].x = Kt[buf][m * LDP + 4 * s + 2 * hf];
            kf[s].y = Kt[buf][m * LDP + 4 * s + 2 * hf + 1];
        }
        v8f sacc = {};
#pragma unroll
        for (int s = 0; s < 16; ++s)
            sacc = __builtin_amdgcn_wmma_f32_16x16x4_f32(false, qa[s], false, kf[s],
                                                         (short)0, sacc, false, false);

        // ---- scale + mask + online softmax (row stats per vgpr index) ----
        const int j = j0 + m;
        const bool jglobal = ((j & (GSTR - 1)) == 0);
#pragma unroll
        for (int v = 0; v < 8; ++v) {
            const int i = qs + v + 8 * hf;
            int dd = i - j; if (dd < 0) dd = -dd;
            const bool allowed = (dd <= WLOC) || jglobal;
            float sv = sacc[v] * scale + (allowed ? 0.0f : -1e30f);

            float r = sv;                               // row max over 16 lanes
            r = fmaxf(r, __shfl_xor(r, 1));
            r = fmaxf(r, __shfl_xor(r, 2));
            r = fmaxf(r, __shfl_xor(r, 4));
            r = fmaxf(r, __shfl_xor(r, 8));
            const float nm = fmaxf(mrow[v], r);
            const float p  = __expf(sv - nm);
            float rs = p;                               // row sum
            rs += __shfl_xor(rs, 1);
            rs += __shfl_xor(rs, 2);
            rs += __shfl_xor(rs, 4);
            rs += __shfl_xor(rs, 8);
            const float alpha = __expf(mrow[v] - nm);
            lrow[v] = lrow[v] * alpha + rs;
            mrow[v] = nm;
            o0[v] *= alpha; o1[v] *= alpha; o2[v] *= alpha; o3[v] *= alpha;
            Pt[(v + 8 * hf) * 17 + m] = p;              // C-layout -> LDS
        }
        __syncthreads();

        // ---- O += P * V (A = P from LDS in A-layout, B = V from LDS) ----
#pragma unroll
        for (int s = 0; s < 4; ++s) {
            const int k0 = 4 * s + 2 * hf;
            v2f pa;
            pa.x = Pt[m * 17 + k0];
            pa.y = Pt[m * 17 + k0 + 1];
#pragma unroll
            for (int g = 0; g < 4; ++g) {
                v2f b;
                b.x = Vt[buf][(k0)     * LDP + g * 16 + m];
                b.y = Vt[buf][(k0 + 1) * LDP + g * 16 + m];
                if (g == 0) o0 = __builtin_amdgcn_wmma_f32_16x16x4_f32(false, pa, false, b, (short)0, o0, false, false);
                if (g == 1) o1 = __builtin_amdgcn_wmma_f32_16x16x4_f32(false, pa, false, b, (short)0, o1, false, false);
                if (g == 2) o2 = __builtin_amdgcn_wmma_f32_16x16x4_f32(false, pa, false, b, (short)0, o2, false, false);
                if (g == 3) o3 = __builtin_amdgcn_wmma_f32_16x16x4_f32(false, pa, false, b, (short)0, o3, false, false);
            }
        }
        __syncthreads();

        buf ^= 1;
        kt = nxt;
    }

    // ---- normalize and store (head-interleaved layout: o[t, h*64 + d]) ----
#pragma unroll
    for (int v = 0; v < 8; ++v) {
        const int row = qs + v + 8 * hf;
        const float inv = 1.0f / lrow[v];
        float* orow = o + (size_t)row * D_MODEL + h * HDIM;
        orow[ 0 + m] = o0[v] * inv;
        orow[16 + m] = o1[v] * inv;
        orow[32 + m] = o2[v] * inv;
        orow[48 + m] = o3[v] * inv;
    }
}

// ---------------------------------------------------------------------------
extern "C" void kernel_launch(void* const* d_in, const int* in_sizes, int n_in,
                              void* d_out, int out_size, void* d_ws, size_t ws_size,
                              hipStream_t stream) {
    const float* x     = (const float*)d_in[0];
    const float* in_w  = (const float*)d_in[1];   // (3D x D)
    const float* in_b  = (const float*)d_in[2];   // (3D)
    const float* out_w = (const float*)d_in[3];   // (D x D)
    const float* out_b = (const float*)d_in[4];   // (D)
    float* out = (float*)d_out;

    // workspace: xr (16MB) | qkv (48MB) | o (16MB)  = 80MB
    float* xr   = (float*)d_ws;
    float* qkv  = xr  + (size_t)T_SEQ * D_MODEL;
    float* obuf = qkv + (size_t)T_SEQ * 3 * D_MODEL;

    // 1) RoPE on x
    rope_kernel<<<(T_SEQ * (D_MODEL / 2)) / 256, 256, 0, stream>>>(x, xr);

    // 2) QKV projection: (4096x1024) @ (1024x3072) + b
    gemm_wmma<<<dim3(T_SEQ / 16, (3 * D_MODEL) / 64), 32, 0, stream>>>(
        xr, D_MODEL, in_w, in_b, qkv, 3 * D_MODEL, D_MODEL);

    // 3) sparse flash attention (async double-buffered KV tiles)
    attn_kernel<<<dim3(T_SEQ / 16, NH), 32, 0, stream>>>(qkv, obuf);

    // 4) output projection: (4096x1024) @ (1024x1024) + b
    gemm_wmma<<<dim3(T_SEQ / 16, D_MODEL / 64), 32, 0, stream>>>(
        obuf, D_MODEL, out_w, out_b, out, D_MODEL, D_MODEL);
}